// GATConv_37254546325571
// MI455X (gfx1250) — compile-verified
//
#include <hip/hip_runtime.h>
#include <math.h>

#define HEADS 8
#define DIM 32
#define FDIM 256   // HEADS * DIM
#define NCOL 16    // 8 el cols + 8 er cols
#define NEG_SLOPE 0.2f

typedef __attribute__((ext_vector_type(2))) float v2f;
typedef __attribute__((ext_vector_type(8))) float v8f;

// ---------------------------------------------------------------------------
// Phase 1: eler[n, c] = feat[n, :] @ W[:, c]   (c<8: attn_l head c, c>=8: attn_r)
// W is block structured: W[k, c] nonzero only when k/32 == c (el) or k/32 == c-8 (er).
// One wave computes a 16-node x 16-col tile via V_WMMA_F32_16X16X4_F32, 64 K-steps.
// B is synthesized branchlessly: unconditional cache-hot 8B load + cndmask select.
// ---------------------------------------------------------------------------
__global__ __launch_bounds__(256) void gat_eler_wmma(
    const float* __restrict__ feat,
    const float* __restrict__ attn_l,
    const float* __restrict__ attn_r,
    float* __restrict__ eler,
    int n_nodes)
{
    const int lane   = threadIdx.x & 31;
    const int wid    = threadIdx.x >> 5;
    const int tile   = blockIdx.x * 8 + wid;
    const int node0  = tile * 16;

    const int  c      = lane & 15;       // matrix column (and A row index)
    const bool hiHalf = (lane >= 16);
    const int  koff   = hiHalf ? 2 : 0;  // K sub-offset held by this half-wave

    // A-matrix 16x4 f32 layout (ISA 7.12.2): lanes 0-15 hold K=k+0,k+1 in
    // VGPR0/1; lanes 16-31 hold K=k+2,k+3.  Row = node0 + (lane&15).
    const int row  = node0 + c;
    const int rowc = (row < n_nodes) ? row : (n_nodes - 1);   // clamp tail loads
    const float* frow = feat + (size_t)rowc * FDIM;

    // Per-lane B source: column c reads only head block (c&7) of attn_l (c<8)
    // or attn_r (c>=8).  Loads below are always in-bounds within that 32-float
    // block ((k&31)+koff+1 <= 31), so no branching is ever needed.
    const int hb = c & 7;
    const float* bptr = ((c < 8) ? attn_l : attn_r) + hb * DIM + koff;

    v8f acc = {};
    #pragma unroll 8
    for (int k = 0; k < FDIM; k += 4) {
        v2f a = *(const v2f*)(frow + k + koff);          // 8B aligned

        v2f braw = *(const v2f*)(bptr + (k & 31));       // L0-hot, 8B aligned
        const bool inr = ((k >> 5) == hb);               // this K-step hits our block
        v2f b;
        b.x = inr ? braw.x : 0.0f;                       // v_cndmask, no divergence
        b.y = inr ? braw.y : 0.0f;

        acc = __builtin_amdgcn_wmma_f32_16x16x4_f32(
                  false, a, false, b, (short)0, acc, false, false);
    }

    // C/D 16x16 f32 layout: lanes 0-15 -> rows j, lanes 16-31 -> rows j+8.
    #pragma unroll
    for (int j = 0; j < 8; ++j) {
        int orow = node0 + j + (hiHalf ? 8 : 0);
        if (orow < n_nodes) eler[orow * NCOL + c] = acc[j];
    }
}

// ---------------------------------------------------------------------------
// Phase 2: one wave per destination node.  dst_idx is sorted -> contiguous edge
// run found by binary search.  Online (flash-style) segment softmax + weighted
// aggregation in one pass: no atomics, no extra edge passes.
// Lane = feature dim d (0..31); lanes 0..7 also carry the 8 head scores.
// ---------------------------------------------------------------------------
__global__ __launch_bounds__(256) void gat_aggregate(
    const float* __restrict__ feat,
    const int*   __restrict__ src_idx,
    const int*   __restrict__ dst_idx,
    const float* __restrict__ eler,
    float*       __restrict__ out,
    int n_nodes, int n_edges)
{
    const int lane = threadIdx.x & 31;
    const int v    = blockIdx.x * 8 + (threadIdx.x >> 5);
    if (v >= n_nodes) return;

    // lower_bound(v) .. lower_bound(v+1) over sorted dst_idx (uniform per wave)
    int lo = 0, hi = n_edges;
    while (lo < hi) { int mid = (lo + hi) >> 1; if (dst_idx[mid] < v) lo = mid + 1; else hi = mid; }
    const int e_begin = lo;
    hi = n_edges;
    while (lo < hi) { int mid = (lo + hi) >> 1; if (dst_idx[mid] < v + 1) lo = mid + 1; else hi = mid; }
    const int e_end = lo;

    const float er_v = eler[v * NCOL + 8 + (lane & 7)];

    float m[HEADS], s[HEADS], acc[HEADS];
    #pragma unroll
    for (int h = 0; h < HEADS; ++h) { m[h] = -INFINITY; s[h] = 0.0f; acc[h] = 0.0f; }

    for (int e = e_begin; e < e_end; ++e) {
        const int src = src_idx[e];
        const float* frow = feat + (size_t)src * FDIM;

        // prefetch next edge's feature row into cache (global_prefetch_b8)
        if (e + 1 < e_end) {
            const int srcn = src_idx[e + 1];
            __builtin_prefetch(feat + (size_t)srcn * FDIM + lane, 0, 3);
        }

        // lanes 0..7 compute the 8 head scores (others compute duplicates)
        float x      = eler[src * NCOL + (lane & 7)] + er_v;
        float escore = (x > 0.0f) ? x : NEG_SLOPE * x;

        #pragma unroll
        for (int h = 0; h < HEADS; ++h) {
            float f    = frow[h * DIM + lane];
            float eh   = __shfl(escore, h, 32);
            float mn   = fmaxf(m[h], eh);
            float corr = __expf(m[h] - mn);   // first edge: exp(-inf)=0 zeroes history
            float p    = __expf(eh - mn);
            s[h]   = s[h]   * corr + p;
            acc[h] = acc[h] * corr + p * f;
            m[h]   = mn;
        }
    }

    float* orow = out + (size_t)v * FDIM;
    #pragma unroll
    for (int h = 0; h < HEADS; ++h) {
        // isolated node (no incoming edges): reference produces 0 via segment_sum
        float val = (s[h] > 0.0f) ? (acc[h] / s[h]) : 0.0f;
        orow[h * DIM + lane] = val;
    }
}

// ---------------------------------------------------------------------------
extern "C" void kernel_launch(void* const* d_in, const int* in_sizes, int n_in,
                              void* d_out, int out_size, void* d_ws, size_t ws_size,
                              hipStream_t stream)
{
    const float* feat   = (const float*)d_in[0];
    const float* attn_l = (const float*)d_in[1];
    const float* attn_r = (const float*)d_in[2];
    const int*   src_idx = (const int*)d_in[3];
    const int*   dst_idx = (const int*)d_in[4];
    float*       out    = (float*)d_out;

    const int n_nodes = in_sizes[0] / FDIM;
    const int n_edges = in_sizes[3];

    float* eler = (float*)d_ws;   // n_nodes * 16 floats (~3.2 MB)

    const int tiles   = (n_nodes + 15) / 16;
    const int blocks1 = (tiles + 7) / 8;            // 8 waves (tiles) per block
    gat_eler_wmma<<<blocks1, 256, 0, stream>>>(feat, attn_l, attn_r, eler, n_nodes);

    const int blocks2 = (n_nodes + 7) / 8;          // 8 nodes (waves) per block
    gat_aggregate<<<blocks2, 256, 0, stream>>>(feat, src_idx, dst_idx, eler, out,
                                               n_nodes, n_edges);
}